// HopfRNNLayerTheta_80221399154861
// MI455X (gfx1250) — compile-verified
//
#include <hip/hip_runtime.h>
#include <math.h>

// HopfRNNLayerTheta for MI455X (gfx1250).
//   x: (64, 2048, 511) f32; A,Bw: (64,64) c64; state0: (64,512) c64
//   out: (64, 2048, 512) f32
// Pipeline: rfft(2048->512 bins)/S -> 512-step complex RNN scan (f32 WMMA,
//           async-DMA double-buffered inputs) -> irfft(512 bins -> 2048).
// Requires ws_size >= 256 MiB:
//   Xf = float2[512 u][512 f(pad)][64 b]   (134,217,728 B)
//   Y  = float2[64 b][512 f(pad)][512 u]   (134,217,728 B)

#define BZ    64      // recurrent state dimension (dim-0 of reference arrays)
#define SEQL  2048
#define FEAT  511
#define UNITS 512
#define NFP   512     // padded feature count (feature 511 = zero pad)
#define NU    512     // kept rfft bins == scan steps

typedef float v2f __attribute__((ext_vector_type(2)));
typedef float v8f __attribute__((ext_vector_type(8)));

#define PIf 3.14159265358979323846f

// ---------------- Stockham radix-2 complex FFT, 1024 points, 256 threads ---
// SIGN = -1 forward (e^{-i}), +1 inverse (e^{+i}); unnormalized.
// Result ends back in (xr, xi) after the 10 (even) ping-pong stages.
template<int SIGN>
__device__ inline void fft1024(float* xr, float* xi, float* yr, float* yi, int t)
{
  float *ar = xr, *ai = xi, *br = yr, *bi = yi;
  int n = 1024, s = 1;
  while (n > 1) {
    const int m = n >> 1;
    const float theta = (float)SIGN * (2.0f * PIf) / (float)n;
    for (int j = t; j < 512; j += 256) {
      const int p = j / s;
      const int q = j - p * s;
      float sn, cs;
      sincosf(theta * (float)p, &sn, &cs);
      const float aRe = ar[q + s * p],       aIm = ai[q + s * p];
      const float bRe = ar[q + s * (p + m)], bIm = ai[q + s * (p + m)];
      br[q + s * 2 * p]       = aRe + bRe;
      bi[q + s * 2 * p]       = aIm + bIm;
      const float dRe = aRe - bRe, dIm = aIm - bIm;
      br[q + s * (2 * p + 1)] = cs * dRe - sn * dIm;
      bi[q + s * (2 * p + 1)] = cs * dIm + sn * dRe;
    }
    __syncthreads();
    float* tr = ar; ar = br; br = tr;
    float* ti = ai; ai = bi; bi = ti;
    n = m; s <<= 1;
  }
}

// ---------------- forward: x[b, :, f] -> Xf[u][f][b] = rfft(x)/S, u<512 -----
__global__ void __launch_bounds__(256) fwd_fft_kernel(const float* __restrict__ x,
                                                      float2* __restrict__ Xf)
{
  __shared__ float sxr[1024], sxi[1024], syr[1024], syi[1024];
  const int f = blockIdx.x % FEAT;
  const int b = blockIdx.x / FEAT;
  const int t = threadIdx.x;
  const float* src = x + (size_t)b * SEQL * FEAT + f;
  for (int i = t; i < 1024; i += 256) {        // pack even/odd into complex
    sxr[i] = src[(size_t)(2 * i)     * FEAT];
    sxi[i] = src[(size_t)(2 * i + 1) * FEAT];
  }
  __syncthreads();
  fft1024<-1>(sxr, sxi, syr, syi, t);
  // real-FFT untwiddle: X[k] = E + e^{-i pi k/1024} * O, keep k = 0..511
  for (int k = t; k < NU; k += 256) {
    const int km = (1024 - k) & 1023;
    const float zr = sxr[k],  zi = sxi[k];
    const float wr = sxr[km], wi = -sxi[km];           // conj(Z[1024-k])
    const float Er = 0.5f * (zr + wr), Ei = 0.5f * (zi + wi);
    const float Dr = 0.5f * (zr - wr), Di = 0.5f * (zi - wi);
    const float Or = Di, Oi = -Dr;                     // O = -i * D
    float sn, cs;
    sincosf(-PIf * (float)k / 1024.0f, &sn, &cs);
    const float Xr = Er + cs * Or - sn * Oi;
    const float Xi = Ei + cs * Oi + sn * Or;
    const float inv = 1.0f / 2048.0f;                  // the "/S" in reference
    Xf[((size_t)k * NFP + f) * BZ + b] = make_float2(Xr * inv, Xi * inv);
  }
}

// ---------------- complex tanh: (sinh2a + i sin2b) / (cosh2a + cos2b) ------
__device__ inline void ctanhf_dev(float a, float b, float& tr, float& ti)
{
  const float x2 = 2.0f * a, y2 = 2.0f * b;
  if (fabsf(x2) > 30.0f) {
    tr = (a > 0.0f) ? 1.0f : -1.0f;
    ti = 0.0f;
  } else {
    float sn, cs;
    sincosf(y2, &sn, &cs);
    const float d = coshf(x2) + cs;
    tr = sinhf(x2) / d;
    ti = sn / d;
  }
}

// ---------------- scan: z <- tanh(A z + B x_u), 16 columns per wave --------
// f32 WMMA (V_WMMA_F32_16X16X4_F32): D(16x16) = A(16x4) x B(4x16) + C.
// Complex product as 8 real WMMAs; f32 WMMA has no A/B NEG modifier, so the
// imaginary-part subtractions use VALU-negated fragments (-zi, -xi).
// X tiles are double-buffered in LDS via GLOBAL_LOAD_ASYNC_TO_LDS_B128
// (ASYNCcnt-tracked DMA): step u+1's 8 KB tile streams in while step u's
// WMMA chain runs, hiding global latency on the serial critical path.
#define LDA  66   // float stride for A/B/Z planes: conflict-free b64 loads
#define XST  66   // float2 stride for staged X columns: 528 B, 16-B aligned

// one tile = 16 columns x 64 float2; one B128 async op moves one 512-B column
__device__ inline void async_load_tile(const float2* __restrict__ gsrc,
                                       const float2* stg, int lane)
{
#pragma unroll
  for (int c = 0; c < 16; ++c) {
    const unsigned long long ga =
        (unsigned long long)(const void*)(gsrc + c * 64) + (unsigned)(lane * 16);
    const unsigned int la =
        (unsigned int)(size_t)(stg + c * XST) + (unsigned)(lane * 16);
    asm volatile("global_load_async_to_lds_b128 %0, %1, off"
                 :: "v"(la), "v"(ga) : "memory");
  }
}

__global__ void __launch_bounds__(32) scan_kernel(
    const float2* __restrict__ A, const float2* __restrict__ Bw,
    const float2* __restrict__ st0,
    const float2* __restrict__ Xf, float2* __restrict__ Y)
{
  __shared__ float sAr[64 * LDA], sAi[64 * LDA], sBr[64 * LDA], sBi[64 * LDA];
  __shared__ float sZr[16 * LDA], sZi[16 * LDA];
  __shared__ __align__(16) float2 sXstg[2][16 * XST];  // raw interleaved tiles

  const int lane = threadIdx.x;       // one wave32 per workgroup
  const int f0 = blockIdx.x * 16;     // this wave owns columns f0..f0+15

  for (int i = lane; i < 64 * 64; i += 32) {   // A, Bw row-major into LDS
    const int r = i >> 6, c = i & 63;
    const float2 av = A[i];
    const float2 bv = Bw[i];
    sAr[r * LDA + c] = av.x;  sAi[r * LDA + c] = av.y;
    sBr[r * LDA + c] = bv.x;  sBi[r * LDA + c] = bv.y;
  }
  for (int i = lane; i < 16 * 64; i += 32) {   // Z column-major: [col][k]
    const int col = i >> 6, k = i & 63;
    const int unit = f0 + col + 1;             // column f maps to unit f+1
    float2 z = make_float2(0.0f, 0.0f);
    if (unit < UNITS) z = st0[(size_t)k * UNITS + unit];
    sZr[col * LDA + k] = z.x;
    sZi[col * LDA + k] = z.y;
  }
  // prologue: DMA tile 0 into buffer 0
  async_load_tile(Xf + (size_t)f0 * BZ, &sXstg[0][0], lane);
  __syncthreads();

  for (int u = 0; u < NU; ++u) {
    const int cur = u & 1;
    // prefetch next tile into the other buffer (dummy re-issue on last step
    // keeps the asynccnt wait threshold uniform; in-order completion means
    // "<=16 outstanding" proves the current tile's 16 older ops are done)
    const int un = (u + 1 < NU) ? (u + 1) : u;
    async_load_tile(Xf + ((size_t)un * NFP + f0) * BZ, &sXstg[cur ^ 1][0], lane);
    asm volatile("s_wait_asynccnt 0x10" ::: "memory");

    const float2* stg = &sXstg[cur][0];
    v8f accR[4] = {};
    v8f accI[4] = {};
    const int colL = lane & 15;
    const int hi2  = (lane >> 4) << 1;
    const int bofs_base = colL * LDA + hi2;
#pragma unroll
    for (int k0 = 0; k0 < 64; k0 += 4) {
      const int bofs = bofs_base + k0;
      const v2f zr  = *(const v2f*)&sZr[bofs];
      const v2f zi  = *(const v2f*)&sZi[bofs];
      const v2f zin = -zi;
      const float* sp = (const float*)(stg + colL * XST + k0 + hi2);
      const v2f xr  = { sp[0], sp[2] };       // re parts of k, k+1
      const v2f xi  = { sp[1], sp[3] };       // im parts of k, k+1
      const v2f xin = -xi;
#pragma unroll
      for (int m = 0; m < 4; ++m) {
        const int aofs = (m * 16 + colL) * LDA + k0 + hi2;
        const v2f ar = *(const v2f*)&sAr[aofs];
        const v2f ai = *(const v2f*)&sAi[aofs];
        const v2f br = *(const v2f*)&sBr[aofs];
        const v2f bi = *(const v2f*)&sBi[aofs];
        // Re: Ar*Zr - Ai*Zi + Br*Xr - Bi*Xi
        accR[m] = __builtin_amdgcn_wmma_f32_16x16x4_f32(false, ar, false, zr,  (short)0, accR[m], false, false);
        accR[m] = __builtin_amdgcn_wmma_f32_16x16x4_f32(false, ai, false, zin, (short)0, accR[m], false, false);
        accR[m] = __builtin_amdgcn_wmma_f32_16x16x4_f32(false, br, false, xr,  (short)0, accR[m], false, false);
        accR[m] = __builtin_amdgcn_wmma_f32_16x16x4_f32(false, bi, false, xin, (short)0, accR[m], false, false);
        // Im: Ar*Zi + Ai*Zr + Br*Xi + Bi*Xr
        accI[m] = __builtin_amdgcn_wmma_f32_16x16x4_f32(false, ar, false, zi,  (short)0, accI[m], false, false);
        accI[m] = __builtin_amdgcn_wmma_f32_16x16x4_f32(false, ai, false, zr,  (short)0, accI[m], false, false);
        accI[m] = __builtin_amdgcn_wmma_f32_16x16x4_f32(false, br, false, xi,  (short)0, accI[m], false, false);
        accI[m] = __builtin_amdgcn_wmma_f32_16x16x4_f32(false, bi, false, xr,  (short)0, accI[m], false, false);
      }
    }
    __syncthreads();

    // tanh + state update + spectrum output Y[b][f][u]
#pragma unroll
    for (int m = 0; m < 4; ++m) {
#pragma unroll
      for (int r = 0; r < 8; ++r) {
        const int row = m * 16 + r + ((lane >> 4) << 3);  // C/D: M = r (+8 hi lanes)
        const int col = lane & 15;
        float tr, ti;
        ctanhf_dev(accR[m][r], accI[m][r], tr, ti);
        sZr[col * LDA + row] = tr;
        sZi[col * LDA + row] = ti;
        Y[((size_t)row * NFP + (f0 + col)) * NU + u] = make_float2(tr, ti);
      }
    }
    __syncthreads();
  }
}

// ---------------- inverse: Y[b][f][0..511] -> out[b][:][f+1] ---------------
// reference: irfft(Y * S, n=2048); the S and the 1/N of irfft cancel, so
// x[2n]+i x[2n+1] = sum_k Zy[k] e^{+2pi i k n/1024} (unnormalized), with
// Zy[k] = (Y[k]+conj(Y[1024-k])) + i e^{i pi k/1024} (Y[k]-conj(Y[1024-k])).
__global__ void __launch_bounds__(256) inv_fft_kernel(const float2* __restrict__ Y,
                                                      float* __restrict__ out)
{
  __shared__ float sxr[1024], sxi[1024], syr[1024], syi[1024];
  const int f = blockIdx.x % FEAT;
  const int b = blockIdx.x / FEAT;
  const int t = threadIdx.x;
  const float2* src = Y + ((size_t)b * NFP + f) * NU;
  for (int k = t; k < 1024; k += 256) {
    float zr = 0.0f, zi = 0.0f;
    float sn, cs;
    sincosf(PIf * (float)k / 1024.0f, &sn, &cs);
    if (k < NU) {                       // conj partner bin is zero
      const float2 yv = src[k];
      const float pr = 1.0f - sn, pi = cs;     // 1 + i e^{i pi k/1024}
      zr = yv.x * pr - yv.y * pi;
      zi = yv.x * pi + yv.y * pr;
    } else if (k > NU) {                // direct bin is zero
      const float2 yv = src[1024 - k];
      const float cr = yv.x, ci = -yv.y;       // conj
      const float pr = 1.0f + sn, pi = -cs;    // 1 - i e^{i pi k/1024}
      zr = cr * pr - ci * pi;
      zi = cr * pi + ci * pr;
    }
    sxr[k] = zr;
    sxi[k] = zi;
  }
  __syncthreads();
  fft1024<1>(sxr, sxi, syr, syi, t);
  const int unit = f + 1;
  float* dst = out + (size_t)b * SEQL * UNITS + unit;
  for (int n = t; n < 1024; n += 256) {
    dst[(size_t)(2 * n)     * UNITS] = sxr[n];
    dst[(size_t)(2 * n + 1) * UNITS] = sxi[n];
  }
}

// ---------------- small init kernels ---------------------------------------
__global__ void pad_xf_kernel(float2* __restrict__ Xf) {
  const int i = blockIdx.x * blockDim.x + threadIdx.x;
  if (i < NU * BZ) {
    const int u = i >> 6, b = i & 63;
    Xf[((size_t)u * NFP + FEAT) * BZ + b] = make_float2(0.0f, 0.0f);
  }
}

__global__ void zero_unit0_kernel(float* __restrict__ out) {
  const int i = blockIdx.x * blockDim.x + threadIdx.x;
  if (i < BZ * SEQL) out[(size_t)i * UNITS] = 0.0f;   // out[b][s][0] = 0
}

// ---------------- host side ------------------------------------------------
extern "C" void kernel_launch(void* const* d_in, const int* in_sizes, int n_in,
                              void* d_out, int out_size, void* d_ws, size_t ws_size,
                              hipStream_t stream)
{
  (void)in_sizes; (void)n_in; (void)out_size; (void)ws_size;
  const float*  x   = (const float*)d_in[0];
  const float2* A   = (const float2*)d_in[1];
  const float2* Bw  = (const float2*)d_in[2];
  const float2* st0 = (const float2*)d_in[3];
  float* out = (float*)d_out;

  float2* Xf = (float2*)d_ws;
  float2* Y  = (float2*)((char*)d_ws + (size_t)NU * NFP * BZ * sizeof(float2));

  pad_xf_kernel<<<(NU * BZ + 255) / 256, 256, 0, stream>>>(Xf);
  fwd_fft_kernel<<<BZ * FEAT, 256, 0, stream>>>(x, Xf);
  scan_kernel<<<32, 32, 0, stream>>>(A, Bw, st0, Xf, Y);
  inv_fft_kernel<<<BZ * FEAT, 256, 0, stream>>>(Y, out);
  zero_unit0_kernel<<<(BZ * SEQL + 255) / 256, 256, 0, stream>>>(out);
}